// EnsembleRNNModule_30983894073775
// MI455X (gfx1250) — compile-verified
//
#include <hip/hip_runtime.h>
#include <math.h>

typedef float v2f __attribute__((ext_vector_type(2)));
typedef float v8f __attribute__((ext_vector_type(8)));

#define E_   8
#define BN_  1024
#define T_   32
#define IN_  31
#define NF_  32
#define HPD_ 64
#define PROJ_ 192
#define EMB_ 32
#define NTOT_ 12481   // 6*2048 + 192 + 1

// ---------------------------------------------------------------------------
// K1: h1 = gelu(emb @ W_in^T + b_in)   (E,BN,32) — tiny, one wave per (e,b)
// ---------------------------------------------------------------------------
__global__ __launch_bounds__(256) void k1_h1(const float* __restrict__ emb,
                                             const float* __restrict__ Win,
                                             const float* __restrict__ bin,
                                             float* __restrict__ h1) {
    int gw   = (blockIdx.x * blockDim.x + threadIdx.x) >> 5;  // (e*BN+b)
    int lane = threadIdx.x & 31;                               // o = lane
    if (gw >= E_ * BN_) return;
    int e = gw >> 10;
    const float* er = emb + (size_t)gw * EMB_;
    const float* wr = Win + ((size_t)e * EMB_ + lane) * EMB_;
    float acc = bin[e * EMB_ + lane];
    #pragma unroll
    for (int i = 0; i < EMB_; ++i) acc = fmaf(wr[i], er[i], acc);
    float g = 0.5f * acc * (1.0f + erff(acc * 0.70710678118654752f));  // exact gelu
    h1[(size_t)gw * EMB_ + lane] = g;
}

// ---------------------------------------------------------------------------
// K2: big hypernet GEMM (WMMA f32 16x16x4): off = h1 @ Wout^T + b_out
//     for the 6 x 2048 stacked-weight offset columns. Fused per-row sumsq.
//     grid: (((e*64+mt)*6+j)*16+nb), 256 thr = 8 waves, wave => 16 N-cols
// ---------------------------------------------------------------------------
__global__ __launch_bounds__(256) void k2_off(const float* __restrict__ h1,
                                              const float* __restrict__ Wout,
                                              const float* __restrict__ bout,
                                              float* __restrict__ off,
                                              float* __restrict__ sumsq) {
    __shared__ float rowsq[16];
    int bidx = blockIdx.x;
    int nb = bidx & 15;  bidx >>= 4;
    int j  = bidx % 6;   bidx /= 6;
    int mt = bidx & 63;
    int e  = bidx >> 6;

    int tid  = threadIdx.x;
    int wave = tid >> 5;
    int lane = tid & 31;
    int lrow = lane & 15;
    int lk   = (lane >> 4) * 2;      // K offset for this lane half
    int mrow0 = mt * 16;
    int ncol  = nb * 128 + wave * 16 + lrow;      // column in [0,2048)

    if (tid < 16) rowsq[tid] = 0.f;
    __syncthreads();

    const float* Arow = h1   + ((size_t)e * BN_ + mrow0 + lrow) * EMB_;
    const float* Brow = Wout + ((size_t)e * NTOT_ + j * 2048 + ncol) * EMB_;

    v8f acc = {};
    #pragma unroll
    for (int ks = 0; ks < 8; ++ks) {
        int k = ks * 4 + lk;
        v2f a;  a.x  = Arow[k];  a.y  = Arow[k + 1];
        v2f bb; bb.x = Brow[k];  bb.y = Brow[k + 1];
        acc = __builtin_amdgcn_wmma_f32_16x16x4_f32(false, a, false, bb,
                                                    (short)0, acc, false, false);
    }

    float bias = bout[(size_t)e * NTOT_ + j * 2048 + ncol];
    int mhalf = (lane >> 4) * 8;
    size_t obase = (((size_t)(e * 6 + j) * BN_) + mrow0) * 2048 + ncol;
    #pragma unroll
    for (int r = 0; r < 8; ++r) {
        float v = acc[r] + bias;
        int ml = r + mhalf;
        off[obase + (size_t)ml * 2048] = v;
        atomicAdd(&rowsq[ml], v * v);
    }
    __syncthreads();
    if (tid < 16)
        atomicAdd(&sumsq[(e * 6 + j) * BN_ + mrow0 + tid], rowsq[tid]);
}

// ---------------------------------------------------------------------------
// Kscale: ref-norm of proj_w[j,e] and the constrain scale per (e,j,b)
// grid: 48 blocks (e*6+j)
// ---------------------------------------------------------------------------
__global__ __launch_bounds__(256) void kscale(const float* __restrict__ projw,
                                              const float* __restrict__ sumsq,
                                              float* __restrict__ scalebuf) {
    __shared__ float partial[256];
    int blk = blockIdx.x;
    int e = blk / 6, j = blk % 6;
    int tid = threadIdx.x;
    const float* pw = projw + (size_t)(j * E_ + e) * 2048;
    float ss = 0.f;
    for (int i = tid; i < 2048; i += 256) { float v = pw[i]; ss += v * v; }
    partial[tid] = ss;
    __syncthreads();
    for (int s = 128; s > 0; s >>= 1) {
        if (tid < s) partial[tid] += partial[tid + s];
        __syncthreads();
    }
    float mx = fmaxf(0.5f * sqrtf(partial[0]), 0.01f);
    for (int b = tid; b < BN_; b += 256) {
        int idx = (e * 6 + j) * BN_ + b;
        scalebuf[idx] = fminf(mx / (sqrtf(sumsq[idx]) + 1e-8f), 1.f);
    }
}

// ---------------------------------------------------------------------------
// K2b: output-head parts (192+1 cols): wn_eff, bn_eff.  One wave per (e,b).
// ---------------------------------------------------------------------------
__global__ __launch_bounds__(256) void k2b_head(const float* __restrict__ h1,
                                                const float* __restrict__ Wout,
                                                const float* __restrict__ bout,
                                                const float* __restrict__ weight_n,
                                                const float* __restrict__ bias_n,
                                                float* __restrict__ wn_eff,
                                                float* __restrict__ bn_eff) {
    int gw   = (blockIdx.x * blockDim.x + threadIdx.x) >> 5;
    int lane = threadIdx.x & 31;
    if (gw >= E_ * BN_) return;
    int e = gw >> 10;
    const float* hr = h1 + (size_t)gw * EMB_;

    float offv[6];
    float ss = 0.f, rs = 0.f;
    #pragma unroll
    for (int seg = 0; seg < 6; ++seg) {
        int o = seg * 32 + lane;
        const float* wr = Wout + ((size_t)e * NTOT_ + 12288 + o) * EMB_;
        float d = bout[(size_t)e * NTOT_ + 12288 + o];
        for (int i = 0; i < EMB_; ++i) d = fmaf(wr[i], hr[i], d);
        offv[seg] = d;
        ss += d * d;
        float w = weight_n[e * PROJ_ + o];
        rs += w * w;
    }
    for (int m = 16; m > 0; m >>= 1) {
        ss += __shfl_xor(ss, m, 32);
        rs += __shfl_xor(rs, m, 32);
    }
    float scale = fminf(fmaxf(0.5f * sqrtf(rs), 0.01f) / (sqrtf(ss) + 1e-8f), 1.f);
    #pragma unroll
    for (int seg = 0; seg < 6; ++seg) {
        int o = seg * 32 + lane;
        wn_eff[(size_t)gw * PROJ_ + o] = weight_n[e * PROJ_ + o] + scale * offv[seg];
    }
    if (lane == 0) {
        const float* wr = Wout + ((size_t)e * NTOT_ + 12480) * EMB_;
        float d = bout[(size_t)e * NTOT_ + 12480];
        for (int i = 0; i < EMB_; ++i) d = fmaf(wr[i], hr[i], d);
        float bref = bias_n[e];
        float sb = fminf(fmaxf(0.5f * fabsf(bref), 0.01f) / (fabsf(d) + 1e-8f), 1.f);
        bn_eff[gw] = bref + sb * d;
    }
}

// ---------------------------------------------------------------------------
// K3: per-participant. WMMA computes A_j = W_eff_j[:, :31] @ X  (64x32 per j),
// h-column factored out as c_j (scalar recurrent state). Then T=32 scalar
// recurrence with a 192-wide block reduction per step.
// grid: 8192 blocks (e*BN+b), 256 thr = 8 waves (4 M-tiles x 2 N-tiles)
// ---------------------------------------------------------------------------
__global__ __launch_bounds__(256) void k3_recur(const float* __restrict__ xs,
                                                const float* __restrict__ h0,
                                                const float* __restrict__ projw,
                                                const float* __restrict__ off,
                                                const float* __restrict__ scalebuf,
                                                const float* __restrict__ wn_eff,
                                                const float* __restrict__ bn_eff,
                                                const float* __restrict__ scale_state,
                                                float* __restrict__ ys) {
    __shared__ float A_lds[6 * HPD_ * T_];   // 48 KB
    __shared__ float X_lds[NF_ * T_];        // 4 KB, row 31 zeroed
    __shared__ float c_lds[6 * HPD_];
    __shared__ float wn_lds[PROJ_];
    __shared__ float scl[6];
    __shared__ float red[8];
    __shared__ float h_sh;

    int pb = blockIdx.x;
    int e = pb >> 10, b = pb & 1023;
    int tid = threadIdx.x, wave = tid >> 5, lane = tid & 31;
    int lrow = lane & 15, lk = (lane >> 4) * 2;

    // stage X (feature-major: X_lds[k*32 + t]), pad feature row 31 with zeros
    for (int idx = tid; idx < NF_ * T_; idx += 256) {
        int k = idx >> 5, t = idx & 31;
        X_lds[idx] = (k < IN_) ? xs[(((size_t)t * E_ + e) * BN_ + b) * IN_ + k] : 0.f;
    }
    if (tid < 6) scl[tid] = scalebuf[(e * 6 + tid) * BN_ + b];
    __syncthreads();

    if (tid < 384) {  // c_j[o] = effective weight column 31 (the h feature)
        int j = tid >> 6, o = tid & 63;
        c_lds[tid] = projw[((size_t)(j * E_ + e) * HPD_ + o) * NF_ + 31]
                   + scl[j] * off[(((size_t)(e * 6 + j) * BN_) + b) * 2048 + o * NF_ + 31];
    }
    if (tid < PROJ_) wn_lds[tid] = wn_eff[(size_t)pb * PROJ_ + tid];

    // WMMA: A_j = W_eff_j @ X (K=32; X row 31 == 0 neutralizes the h column)
    int mi = wave >> 1, ni = wave & 1;
    for (int j = 0; j < 6; ++j) {
        float s = scl[j];
        const float* wp = projw + (size_t)(j * E_ + e) * 2048;
        const float* op = off + (((size_t)(e * 6 + j) * BN_) + b) * 2048;
        int o = mi * 16 + lrow;
        v8f acc = {};
        #pragma unroll
        for (int ks = 0; ks < 8; ++ks) {
            int k = ks * 4 + lk;
            v2f a;
            a.x = fmaf(s, op[o * NF_ + k],     wp[o * NF_ + k]);
            a.y = fmaf(s, op[o * NF_ + k + 1], wp[o * NF_ + k + 1]);
            v2f bb;
            bb.x = X_lds[k * T_ + ni * 16 + lrow];
            bb.y = X_lds[(k + 1) * T_ + ni * 16 + lrow];
            acc = __builtin_amdgcn_wmma_f32_16x16x4_f32(false, a, false, bb,
                                                        (short)0, acc, false, false);
        }
        int mhalf = (lane >> 4) * 8;
        #pragma unroll
        for (int r = 0; r < 8; ++r) {
            int orow = mi * 16 + r + mhalf;
            A_lds[j * 2048 + orow * T_ + ni * 16 + lrow] = acc[r];
        }
    }
    __syncthreads();

    // scalar recurrence over T
    float alpha = 1.f / (1.f + __expf(-scale_state[0]));
    float onem = 1.f - alpha;
    float h = h0[pb];
    float bn = bn_eff[pb];
    float wnv = (tid < PROJ_) ? wn_lds[tid] : 0.f;

    for (int t = 0; t < T_; ++t) {
        float val = 0.f;
        if (tid < 64) {
            val = A_lds[tid * T_ + t] + c_lds[tid] * h;
        } else if (tid < 128) {
            int o = tid - 64;
            float m1 = A_lds[1 * 2048 + o * T_ + t] + c_lds[64 + o] * h;
            float m2 = A_lds[2 * 2048 + o * T_ + t] + c_lds[128 + o] * h;
            val = m1 * m2;
        } else if (tid < PROJ_) {
            int o = tid - 128;
            float m3 = A_lds[3 * 2048 + o * T_ + t] + c_lds[192 + o] * h;
            float m4 = A_lds[4 * 2048 + o * T_ + t] + c_lds[256 + o] * h;
            float m5 = A_lds[5 * 2048 + o * T_ + t] + c_lds[320 + o] * h;
            val = m3 * m4 * m5;
        }
        val *= wnv;
        for (int m = 16; m > 0; m >>= 1) val += __shfl_xor(val, m, 32);
        if (lane == 0) red[wave] = val;
        __syncthreads();
        if (tid == 0) {
            float n = red[0] + red[1] + red[2] + red[3] + red[4] + red[5] + bn;
            float hn = onem * h + n;
            h_sh = hn;
            ys[((size_t)t * E_ + e) * BN_ + b] = hn;
        }
        __syncthreads();
        h = h_sh;
    }
}

// ---------------------------------------------------------------------------
extern "C" void kernel_launch(void* const* d_in, const int* in_sizes, int n_in,
                              void* d_out, int out_size, void* d_ws, size_t ws_size,
                              hipStream_t stream) {
    (void)in_sizes; (void)n_in; (void)out_size; (void)ws_size;
    const float* xs          = (const float*)d_in[0];
    const float* emb         = (const float*)d_in[1];
    const float* h0          = (const float*)d_in[2];
    const float* projw       = (const float*)d_in[3];
    const float* weight_n    = (const float*)d_in[4];
    const float* bias_n      = (const float*)d_in[5];
    const float* scale_state = (const float*)d_in[6];
    const float* Win         = (const float*)d_in[7];
    const float* bin         = (const float*)d_in[8];
    const float* Wout        = (const float*)d_in[9];
    const float* bout        = (const float*)d_in[10];
    float* ys = (float*)d_out;

    // scratch layout (f32): h1 | sumsq | scale | wn_eff | bn_eff | off
    float* h1       = (float*)d_ws;                       // 262144
    float* sumsq    = h1 + (size_t)E_ * BN_ * EMB_;       // 49152
    float* scalebuf = sumsq + (size_t)E_ * 6 * BN_;       // 49152
    float* wn_eff   = scalebuf + (size_t)E_ * 6 * BN_;    // 1572864
    float* bn_eff   = wn_eff + (size_t)E_ * BN_ * PROJ_;  // 8192
    float* off      = bn_eff + (size_t)E_ * BN_;          // 100663296

    hipMemsetAsync(sumsq, 0, (size_t)E_ * 6 * BN_ * sizeof(float), stream);

    k1_h1<<<E_ * BN_ / 8, 256, 0, stream>>>(emb, Win, bin, h1);
    k2_off<<<E_ * 64 * 6 * 16, 256, 0, stream>>>(h1, Wout, bout, off, sumsq);
    kscale<<<E_ * 6, 256, 0, stream>>>(projw, sumsq, scalebuf);
    k2b_head<<<E_ * BN_ / 8, 256, 0, stream>>>(h1, Wout, bout, weight_n, bias_n,
                                               wn_eff, bn_eff);
    k3_recur<<<E_ * BN_, 256, 0, stream>>>(xs, h0, projw, off, scalebuf,
                                           wn_eff, bn_eff, scale_state, ys);
}